// LogGD_5377299054915
// MI455X (gfx1250) — compile-verified
//
#include <hip/hip_runtime.h>
#include <hip/hip_bf16.h>

typedef __attribute__((ext_vector_type(2))) float v2f;
typedef __attribute__((ext_vector_type(8))) float v8f;

#define D 64

// ---------------------------------------------------------------------------
// Scatter-add: one wave (32 lanes) per edge; lane l handles features 2l, 2l+1.
// aggr[dst] += x[src]  via hardware f32 global atomics (resolve in L2).
// ---------------------------------------------------------------------------
__global__ void rgcn_scatter_kernel(const float* __restrict__ X,
                                    const int* __restrict__ src,
                                    const int* __restrict__ dst,
                                    float* __restrict__ A, int E) {
    int gid  = blockIdx.x * blockDim.x + threadIdx.x;
    int edge = gid >> 5;
    int lane = threadIdx.x & 31;
    if (edge >= E) return;
    int s = src[edge];
    int d = dst[edge];
    v2f v = *(const v2f*)(X + (size_t)s * D + lane * 2);
    float* out = A + (size_t)d * D + lane * 2;
    unsafeAtomicAdd(out + 0, v.x);
    unsafeAtomicAdd(out + 1, v.y);
}

// ---------------------------------------------------------------------------
// Fused dual-GEMM + bias + ReLU using V_WMMA_F32_16X16X4_F32 (full fp32):
//   H = relu(Aggr @ Wrel + X @ Wroot + b)
// One wave per 16x16 output tile; 4 waves/block cover the 64 output columns
// of one 16-row tile. Optionally skips the store and instead accumulates
// per-column sums (fused mean-pool for the final layer).
//
// Fragment layouts (ISA 7.12.2, 32-bit, wave32):
//   A 16x4 : lane l (m=l&15, kh=2*(l>>4)) holds {A[m][k+kh], A[m][k+kh+1]}
//   B 4x16 : lane l (n=l&15, kh=2*(l>>4)) holds {B[k+kh][n], B[k+kh+1][n]}
//   C 16x16: vgpr r, lane l -> C[r + 8*(l>>4)][l&15]
// ---------------------------------------------------------------------------
__global__ void rgcn_gemm_relu_kernel(const float* __restrict__ Aggr,
                                      const float* __restrict__ X,
                                      const float* __restrict__ Wrel,
                                      const float* __restrict__ Wroot,
                                      const float* __restrict__ bias,
                                      float* __restrict__ Hout,     // may be null
                                      float* __restrict__ colsum,   // may be null
                                      int nRows) {
    int lane  = threadIdx.x & 31;
    int wv    = threadIdx.x >> 5;       // 0..3 -> column tile
    int row0  = blockIdx.x * 16;
    int col0  = wv * 16;
    int m     = lane & 15;               // row-in-tile for A, col for B/C
    int kh    = (lane >> 4) * 2;         // 0 or 2

    const float* arow = Aggr + (size_t)(row0 + m) * D;
    const float* xrow = X    + (size_t)(row0 + m) * D;

    v8f acc = {};
#pragma unroll
    for (int k = 0; k < D; k += 4) {
        v2f a = *(const v2f*)(arow + k + kh);
        v2f b;
        b.x = Wrel[(k + kh)     * D + col0 + m];
        b.y = Wrel[(k + kh + 1) * D + col0 + m];
        acc = __builtin_amdgcn_wmma_f32_16x16x4_f32(
            false, a, false, b, (short)0, acc, false, false);
    }
#pragma unroll
    for (int k = 0; k < D; k += 4) {
        v2f a = *(const v2f*)(xrow + k + kh);
        v2f b;
        b.x = Wroot[(k + kh)     * D + col0 + m];
        b.y = Wroot[(k + kh + 1) * D + col0 + m];
        acc = __builtin_amdgcn_wmma_f32_16x16x4_f32(
            false, a, false, b, (short)0, acc, false, false);
    }

    float bv = bias[col0 + m];
    float csum = 0.0f;
#pragma unroll
    for (int r = 0; r < 8; ++r) {
        int M = r + 8 * (lane >> 4);
        int row = row0 + M;
        float v = acc[r] + bv;
        v = v > 0.0f ? v : 0.0f;
        if (row < nRows) {
            if (Hout) Hout[(size_t)row * D + col0 + m] = v;
            csum += v;
        }
    }
    if (colsum) unsafeAtomicAdd(&colsum[col0 + m], csum);
}

// ---------------------------------------------------------------------------
// out[c] = (colsum/N) @ Wc[:,c] + bc[c],  c = 0..1
// ---------------------------------------------------------------------------
__global__ void rgcn_head_kernel(const float* __restrict__ colsum,
                                 const float* __restrict__ Wc,
                                 const float* __restrict__ bc,
                                 float* __restrict__ out, int nNodes) {
    int c = threadIdx.x;
    if (c < 2) {
        float inv = 1.0f / (float)nNodes;
        float s = 0.0f;
        for (int d = 0; d < D; ++d)
            s += (colsum[d] * inv) * Wc[d * 2 + c];
        out[c] = s + bc[c];
    }
}

extern "C" void kernel_launch(void* const* d_in, const int* in_sizes, int n_in,
                              void* d_out, int out_size, void* d_ws, size_t ws_size,
                              hipStream_t stream) {
    const float* x      = (const float*)d_in[0];
    const int*   eidx   = (const int*)d_in[1];
    const float* W1_rel = (const float*)d_in[2];
    const float* W1_root= (const float*)d_in[3];
    const float* b1     = (const float*)d_in[4];
    const float* W2_rel = (const float*)d_in[5];
    const float* W2_root= (const float*)d_in[6];
    const float* b2     = (const float*)d_in[7];
    const float* Wc     = (const float*)d_in[8];
    const float* bc     = (const float*)d_in[9];
    float* out = (float*)d_out;

    const int N = in_sizes[0] / D;        // 100000
    const int E = in_sizes[1] / 2;        // 1200000
    const int* src = eidx;
    const int* dst = eidx + E;

    float* A      = (float*)d_ws;                 // N*D aggregation buffer
    float* H1     = A + (size_t)N * D;            // N*D hidden layer 1
    float* colsum = H1 + (size_t)N * D;           // D column sums

    const int rowTiles   = (N + 15) / 16;
    const int scatterBlk = (E * 32 + 255) / 256;

    // Layer 1: zero aggr + colsum, scatter x, GEMM -> H1
    hipMemsetAsync(A, 0, (size_t)N * D * sizeof(float), stream);
    hipMemsetAsync(colsum, 0, D * sizeof(float), stream);
    rgcn_scatter_kernel<<<scatterBlk, 256, 0, stream>>>(x, src, dst, A, E);
    rgcn_gemm_relu_kernel<<<rowTiles, 128, 0, stream>>>(
        A, x, W1_rel, W1_root, b1, H1, nullptr, N);

    // Layer 2: zero aggr, scatter H1, GEMM with fused column-sum (no H2 store)
    hipMemsetAsync(A, 0, (size_t)N * D * sizeof(float), stream);
    rgcn_scatter_kernel<<<scatterBlk, 256, 0, stream>>>(H1, src, dst, A, E);
    rgcn_gemm_relu_kernel<<<rowTiles, 128, 0, stream>>>(
        A, H1, W2_rel, W2_root, b2, nullptr, colsum, N);

    // Head
    rgcn_head_kernel<<<1, 32, 0, stream>>>(colsum, Wc, bc, out, N);
}